// ConcatLayer_60567628808508
// MI455X (gfx1250) — compile-verified
//
#include <hip/hip_runtime.h>
#include <stdint.h>

#define TPB            256
#define ROWS_PER_BLK   256                   // rows per block tile
#define IN_DW          (ROWS_PER_BLK * 9)    // 2304 dwords in  (576 float4)
#define OUT_DW         (ROWS_PER_BLK * 3)    // 768 dwords out  (192 float4)

// Reference's _get_max_index on a 3-vector:
//   unique max at index i -> 1 - i  (i.e. {1, 0, -1}); ties -> 0
__device__ __forceinline__ int max_idx3(float a, float b, float c) {
    float m = fmaxf(a, fmaxf(b, c));
    int e0 = (a == m) ? 1 : 0;
    int e1 = (b == m) ? 1 : 0;
    int e2 = (c == m) ? 1 : 0;
    int last = e2 ? 2 : (e1 ? 1 : 0);        // last index holding the max
    return ((e0 + e1 + e2) == 1) ? (1 - last) : 0;
}

__device__ __forceinline__ void concat_row(const float v[9], float r[3]) {
    const float u0 = v[0], u1 = v[1], u2 = v[2];
    const float n0 = v[3], n1 = v[4], n2 = v[5];
    const float d0 = v[6], d1 = v[7], d2 = v[8];

    const int mu = max_idx3(u0, u1, u2);
    const int mn = max_idx3(n0, n1, n2);
    const int md = max_idx3(d0, d1, d2);

    const int calc = (mn < 0 ? -mn : mn) * (mu + md + mn);
    const int sgn  = (calc > 0) - (calc < 0);

    const float fu = (sgn == mu) ? 1.0f : 0.0f;
    const float fn = (sgn == mn) ? 1.0f : 0.0f;
    const float fd = (sgn == md) ? 1.0f : 0.0f;

    const float U0 = u0 * fu, U1 = u1 * fu, U2 = u2 * fu;
    const float N0 = n0 * fn, N1 = n1 * fn, N2 = n2 * fn;
    const float D0 = d0 * fd, D1 = d1 * fd, D2 = d2 * fd;

    const int col = (calc == 0) ? 1 : ((calc == 1) ? 0 : 2);
    const float s0 = (col == 0) ? U0 : ((col == 1) ? U1 : U2);
    const float s1 = (col == 0) ? N0 : ((col == 1) ? N1 : N2);
    const float s2 = (col == 0) ? D0 : ((col == 1) ? D1 : D2);

    const float mm = fmaxf(s0, fmaxf(s1, s2));
    if (s0 == mm)       { r[0] = U0; r[1] = U1; r[2] = U2; }  // first-max wins
    else if (s1 == mm)  { r[0] = N0; r[1] = N1; r[2] = N2; }
    else                { r[0] = D0; r[1] = D1; r[2] = D2; }
}

__global__ __launch_bounds__(TPB)
void ConcatLayer_60567628808508_kernel(const float* __restrict__ x,
                                       float* __restrict__ out, int n_rows) {
    __shared__ __align__(16) float sIn[IN_DW];
    __shared__ __align__(16) float sOut[OUT_DW];

    const int tid      = threadIdx.x;
    const int rowBase  = blockIdx.x * ROWS_PER_BLK;
    const int rowsHere = min(ROWS_PER_BLK, n_rows - rowBase);

    if (rowsHere == ROWS_PER_BLK) {
        // ---- fast path: async-DMA the whole 9216B tile into LDS (all b128) ----
        const float* gIn = x + (size_t)rowBase * 9;           // 9216B-multiple offset: 16B aligned
        const unsigned ldsInBase  = (unsigned)(uintptr_t)&sIn[0];   // low 32 bits of flat LDS ptr = LDS offset
        const unsigned ldsOutBase = (unsigned)(uintptr_t)&sOut[0];

        // 576 float4 = 2 full rounds of 256 lanes + 1 round on lanes 0..63
        {
            unsigned vo0 = (unsigned)tid * 16u;               // bytes
            unsigned vo1 = (unsigned)(tid + 256) * 16u;
            unsigned lo0 = ldsInBase + vo0;
            unsigned lo1 = ldsInBase + vo1;
            asm volatile("global_load_async_to_lds_b128 %0, %1, %2"
                         :: "v"(lo0), "v"(vo0), "s"(gIn) : "memory");
            asm volatile("global_load_async_to_lds_b128 %0, %1, %2"
                         :: "v"(lo1), "v"(vo1), "s"(gIn) : "memory");
            if (tid < 64) {
                unsigned vo2 = (unsigned)(tid + 512) * 16u;
                unsigned lo2 = ldsInBase + vo2;
                asm volatile("global_load_async_to_lds_b128 %0, %1, %2"
                             :: "v"(lo2), "v"(vo2), "s"(gIn) : "memory");
            }
        }
        asm volatile("s_wait_asynccnt 0" ::: "memory");
        __syncthreads();   // every wave passed its own asynccnt wait -> tile fully in LDS

        float v[9], r[3];
        #pragma unroll
        for (int j = 0; j < 9; ++j) v[j] = sIn[tid * 9 + j];    // stride 9: bank-conflict-free
        concat_row(v, r);
        #pragma unroll
        for (int j = 0; j < 3; ++j) sOut[tid * 3 + j] = r[j];   // stride 3: bank-conflict-free

        __syncthreads();   // sOut complete before the async engine reads LDS

        // ---- async-DMA the 3072B result tile out: 192 b128 stores on lanes 0..191 ----
        float* gOut = out + (size_t)rowBase * 3;              // 3072B-multiple offset: 16B aligned
        if (tid < 192) {
            unsigned vo = (unsigned)tid * 16u;                // bytes
            unsigned lo = ldsOutBase + vo;
            asm volatile("global_store_async_from_lds_b128 %0, %1, %2"
                         :: "v"(vo), "v"(lo), "s"(gOut) : "memory");
        }
        asm volatile("s_wait_asynccnt 0" ::: "memory");
    } else {
        // ---- tail path: guarded scalar row ----
        const int row = rowBase + tid;
        if (row < n_rows) {
            float v[9], r[3];
            #pragma unroll
            for (int j = 0; j < 9; ++j) v[j] = x[(size_t)row * 9 + j];
            concat_row(v, r);
            #pragma unroll
            for (int j = 0; j < 3; ++j) out[(size_t)row * 3 + j] = r[j];
        }
    }
}

extern "C" void kernel_launch(void* const* d_in, const int* in_sizes, int n_in,
                              void* d_out, int out_size, void* d_ws, size_t ws_size,
                              hipStream_t stream) {
    (void)n_in; (void)out_size; (void)d_ws; (void)ws_size;
    const float* x   = (const float*)d_in[0];
    float*       out = (float*)d_out;
    const int n_rows = in_sizes[0] / 9;
    const int blocks = (n_rows + ROWS_PER_BLK - 1) / ROWS_PER_BLK;
    ConcatLayer_60567628808508_kernel<<<blocks, TPB, 0, stream>>>(x, out, n_rows);
}